// ModelBuilder_5377299055049
// MI455X (gfx1250) — compile-verified
//
#include <hip/hip_runtime.h>
#include <stdint.h>

// CenterNet decode for MI455X (gfx1250, wave32).
// 3 streaming NMS passes staged through LDS via the Tensor Data Mover,
// two-level radix select on float bits for exact ordered top-100.

#define BATCH 32
#define H     128
#define W     128
#define CHT   84      // channels in detections
#define CCLS  80      // heatmap classes
#define KTOP  100
#define TY    16      // rows per tile (halo overhead (TY+2)/TY = 1.125)
#define CG    8       // channels per tile
#define NTHR  256
#define NBIN  4096
#define CAP   2048

typedef uint32_t v4u __attribute__((ext_vector_type(4)));
typedef int      v8i __attribute__((ext_vector_type(8)));
typedef int      v4i __attribute__((ext_vector_type(4)));

#if defined(__has_builtin)
#if __has_builtin(__builtin_amdgcn_tensor_load_to_lds) && __has_builtin(__builtin_amdgcn_s_wait_tensorcnt)
#define USE_TDM 1
#endif
#endif

// Stage a (TY+2) x W x CG tile of `det` into LDS (halo rows clamped, missing
// halo filled with -inf), then invoke emit(bits, peak, flat_idx) for each of
// the TY*W*CG peak values this block owns.
template <typename F>
__device__ __forceinline__ void nms_pass(const float* __restrict__ det, F emit) {
    __shared__ float tile[TY + 2][W][CG];

    const int cg = blockIdx.x;          // 0..9   channel group
    const int yt = blockIdx.y;          // 0..7   row tile
    const int b  = blockIdx.z;          // 0..31  batch
    const int c0 = cg * CG;
    const int y0 = yt * TY;

    const int ylo   = (y0 - 1 < 0) ? 0 : y0 - 1;
    const int yhi   = (y0 + TY > H - 1) ? H - 1 : y0 + TY;
    const int nrows = yhi - ylo + 1;
    const int skip  = ylo - (y0 - 1);   // 0 or 1

    // Fill uncovered halo rows with -inf (exact SAME padding semantics).
    for (int r = 0; r < TY + 2; ++r) {
        if (r < skip || r >= skip + nrows) {
            float* p = &tile[r][0][0];
            for (int i = threadIdx.x; i < W * CG; i += NTHR) p[i] = -__builtin_inff();
        }
    }

    const float* gsrc = det + (((size_t)(b * H + ylo)) * W) * CHT + c0;

#if defined(USE_TDM)
    if (threadIdx.x < 32) {             // wave 0 issues the TDM (EXEC ignored)
        const uint64_t ga  = (uint64_t)(uintptr_t)gsrc;
        const uint32_t lds = (uint32_t)(uintptr_t)&tile[skip][0][0];
        const uint64_t s0  = CHT;               // dim0 stride: next column
        const uint64_t s1  = (uint64_t)CHT * W; // dim1 stride: next row

        v4u g0;
        g0[0] = 1u;                                          // count=1, normal load
        g0[1] = lds;                                         // lds_addr
        g0[2] = (uint32_t)ga;                                // global_addr[31:0]
        g0[3] = (uint32_t)((ga >> 32) & 0x1FFFFFFu) | (2u << 30); // addr[56:32] | type=2

        v8i g1;
        g1[0] = (int)(2u << 16);                             // data_size = 4 bytes
        g1[1] = (int)(((uint32_t)CHT & 0xFFFFu) << 16);      // tensor_dim0[15:0]
        g1[2] = (int)(((uint32_t)CHT >> 16) | (((uint32_t)W & 0xFFFFu) << 16)); // dim0 hi | dim1 lo
        g1[3] = (int)(((uint32_t)W >> 16) | ((uint32_t)CG << 16));              // dim1 hi | tile_dim0
        g1[4] = (int)(((uint32_t)W & 0xFFFFu) | ((uint32_t)nrows << 16));       // tile_dim1 | tile_dim2
        g1[5] = (int)(uint32_t)(s0 & 0xFFFFFFFFu);           // tensor_dim0_stride[31:0]
        g1[6] = (int)((uint32_t)(s0 >> 32) | ((uint32_t)(s1 & 0xFFFFu) << 16)); // s0 hi | s1 lo
        g1[7] = (int)(uint32_t)(s1 >> 16);                   // tensor_dim1_stride[47:16]

        v4i g2; g2[0] = nrows; g2[1] = 0; g2[2] = 0; g2[3] = 0; // tensor_dim2 = nrows
        v4i g3; g3[0] = 0;     g3[1] = 0; g3[2] = 0; g3[3] = 0;
        v8i g4; g4[0] = 0; g4[1] = 0; g4[2] = 0; g4[3] = 0;     // extra operand of the
        g4[4] = 0; g4[5] = 0; g4[6] = 0; g4[7] = 0;             // 6-arg builtin: zeroed

        __builtin_amdgcn_tensor_load_to_lds(g0, g1, g2, g3, g4, 0);
        __builtin_amdgcn_s_wait_tensorcnt((short)0);
    }
#else
    for (int i = threadIdx.x; i < nrows * W * CG; i += NTHR) {
        int r   = i / (W * CG);
        int rem = i % (W * CG);
        int x   = rem / CG;
        int c   = rem % CG;
        tile[skip + r][x][c] = det[(((size_t)(b * H + ylo + r)) * W + x) * CHT + c0 + c];
    }
#endif
    __syncthreads();

    for (int i = threadIdx.x; i < TY * W * CG; i += NTHR) {
        const int c  = i % CG;
        const int x  = (i / CG) % W;
        const int ty = i / (CG * W);
        const int y  = y0 + ty;
        const int r  = ty + 1;                // tile row of global row y

        // Clamped x-neighbors: duplicate reads are idempotent under max,
        // keeps the 3x3 window branch-free and fully unrolled.
        const int xm = (x - 1 < 0) ? 0 : x - 1;
        const int xp = (x + 1 > W - 1) ? W - 1 : x + 1;

        const float h = tile[r][x][c];
        float m = h;
        #pragma unroll
        for (int rr = -1; rr <= 1; ++rr) {
            m = fmaxf(m, tile[r + rr][xm][c]);
            m = fmaxf(m, tile[r + rr][x ][c]);
            m = fmaxf(m, tile[r + rr][xp][c]);
        }

        const float peak = (fabsf(h - m) < 1e-4f) ? m : 0.0f;
        const unsigned bits = __float_as_uint(peak);
        const int idx = (y * W + x) * CCLS + (c0 + c);
        emit(bits, peak, idx);
    }
}

__global__ void k_init(unsigned* hist1, unsigned* hist2, int* cnt) {
    int i = blockIdx.x * blockDim.x + threadIdx.x;
    if (i < BATCH * NBIN) { hist1[i] = 0u; hist2[i] = 0u; }
    if (i < BATCH) cnt[i] = 0;
}

__global__ void k_hist1(const float* __restrict__ det, unsigned* __restrict__ hist1) {
    __shared__ unsigned lh[NBIN];
    for (int i = threadIdx.x; i < NBIN; i += NTHR) lh[i] = 0u;
    __syncthreads();
    nms_pass(det, [&](unsigned bits, float, int) {
        if (bits) atomicAdd(&lh[bits >> 20], 1u);
    });
    __syncthreads();
    unsigned* gh = hist1 + (size_t)blockIdx.z * NBIN;
    for (int i = threadIdx.x; i < NBIN; i += NTHR)
        if (lh[i]) atomicAdd(&gh[i], lh[i]);
}

__global__ void k_select1(const unsigned* __restrict__ hist1, int* t1, int* above1) {
    int b = threadIdx.x;
    if (b >= BATCH) return;
    const unsigned* hb = hist1 + (size_t)b * NBIN;
    unsigned cum = 0; int tb = 0; unsigned ab = 0; bool found = false;
    for (int bin = NBIN - 1; bin >= 0; --bin) {
        unsigned c = hb[bin];
        if (cum + c >= (unsigned)KTOP) { tb = bin; ab = cum; found = true; break; }
        cum += c;
    }
    if (!found) { tb = 0; ab = cum - hb[0]; }
    t1[b] = tb; above1[b] = (int)ab;
}

__global__ void k_hist2(const float* __restrict__ det, const int* __restrict__ t1,
                        unsigned* __restrict__ hist2) {
    __shared__ unsigned lh[NBIN];
    for (int i = threadIdx.x; i < NBIN; i += NTHR) lh[i] = 0u;
    __syncthreads();
    const unsigned t1g = (unsigned)t1[blockIdx.z];
    nms_pass(det, [&](unsigned bits, float, int) {
        if (bits && (bits >> 20) == t1g) atomicAdd(&lh[(bits >> 8) & 0xFFFu], 1u);
    });
    __syncthreads();
    unsigned* gh = hist2 + (size_t)blockIdx.z * NBIN;
    for (int i = threadIdx.x; i < NBIN; i += NTHR)
        if (lh[i]) atomicAdd(&gh[i], lh[i]);
}

__global__ void k_select2(const unsigned* __restrict__ hist2, const int* __restrict__ t1,
                          const int* __restrict__ above1, unsigned* thresh) {
    int b = threadIdx.x;
    if (b >= BATCH) return;
    const unsigned* hb = hist2 + (size_t)b * NBIN;
    int need = KTOP - above1[b];
    unsigned cum = 0; int t2 = 0;
    for (int bin = NBIN - 1; bin >= 0; --bin) {
        cum += hb[bin];
        if ((int)cum >= need) { t2 = bin; break; }
    }
    thresh[b] = ((unsigned)t1[b] << 20) | ((unsigned)t2 << 8);
}

__global__ void k_collect(const float* __restrict__ det, const unsigned* __restrict__ thresh,
                          int* cnt, float* cs, int* ci) {
    const unsigned th = thresh[blockIdx.z];
    const int b = blockIdx.z;
    nms_pass(det, [&](unsigned bits, float peak, int idx) {
        if (bits && bits >= th) {
            int p = atomicAdd(&cnt[b], 1);
            if (p < CAP) { cs[(size_t)b * CAP + p] = peak; ci[(size_t)b * CAP + p] = idx; }
        }
    });
}

__global__ void k_final(const float* __restrict__ det, const float* __restrict__ cs,
                        const int* __restrict__ ci, const int* __restrict__ cnt,
                        float* __restrict__ out) {
    __shared__ float s[CAP];
    __shared__ int   id[CAP];
    __shared__ float rs[NTHR];
    __shared__ int   rp[NTHR];

    const int b = blockIdx.x;
    const int n = min(cnt[b], CAP);
    for (int i = threadIdx.x; i < n; i += NTHR) {
        s[i]  = cs[(size_t)b * CAP + i];
        id[i] = ci[(size_t)b * CAP + i];
    }
    __syncthreads();

    for (int j = 0; j < KTOP; ++j) {
        // local argmax, tie-break lowest index (matches lax.top_k)
        float bs = -1.0f; int bp = -1;
        for (int i = threadIdx.x; i < n; i += NTHR) {
            float v = s[i];
            if (v > bs || (v == bs && bp >= 0 && id[i] < id[bp])) { bs = v; bp = i; }
        }
        rs[threadIdx.x] = bs; rp[threadIdx.x] = bp;
        __syncthreads();
        for (int off = NTHR / 2; off > 0; off >>= 1) {
            if ((int)threadIdx.x < off) {
                float v1 = rs[threadIdx.x + off]; int p1 = rp[threadIdx.x + off];
                float v0 = rs[threadIdx.x];       int p0 = rp[threadIdx.x];
                bool take = (v1 > v0) ||
                            (v1 == v0 && p1 >= 0 && (p0 < 0 || id[p1] < id[p0]));
                if (take) { rs[threadIdx.x] = v1; rp[threadIdx.x] = p1; }
            }
            __syncthreads();
        }
        if (threadIdx.x == 0) {
            float* o = out + ((size_t)b * KTOP + j) * 6;
            int p = rp[0];
            if (p < 0 || rs[0] <= 0.0f) {
                o[0] = o[1] = o[2] = o[3] = o[4] = o[5] = 0.0f;
            } else {
                float score = s[p];
                int ind = id[p];
                s[p] = -1.0f;                      // remove from future rounds
                int c  = ind % CCLS;
                int sp = ind / CCLS;
                int x  = sp % W;
                int y  = sp / W;
                const float* whp = det + (((size_t)(b * H + y)) * W + x) * CHT + CCLS;
                float ysf = (float)y / (float)H;
                float xsf = (float)x / (float)W;
                o[0] = ysf - whp[0];
                o[1] = xsf - whp[1];
                o[2] = ysf + whp[2];
                o[3] = xsf + whp[3];
                o[4] = (float)c;
                o[5] = score;
            }
        }
        __syncthreads();
    }
}

extern "C" void kernel_launch(void* const* d_in, const int* in_sizes, int n_in,
                              void* d_out, int out_size, void* d_ws, size_t ws_size,
                              hipStream_t stream) {
    const float* det = (const float*)d_in[0];   // (32,128,128,84) f32
    // d_in[1] is k == 100 (compile-time KTOP)
    float* out = (float*)d_out;                 // (32,100,6) f32

    // Workspace layout (~1.6 MB)
    unsigned* hist1  = (unsigned*)d_ws;                       // 32*4096
    unsigned* hist2  = hist1 + BATCH * NBIN;                  // 32*4096
    int*      t1     = (int*)(hist2 + BATCH * NBIN);          // 32
    int*      above1 = t1 + BATCH;                            // 32
    unsigned* thresh = (unsigned*)(above1 + BATCH);           // 32
    int*      cnt    = (int*)(thresh + BATCH);                // 32
    float*    cs     = (float*)(cnt + BATCH);                 // 32*CAP
    int*      ci     = (int*)(cs + BATCH * CAP);              // 32*CAP

    k_init<<<(BATCH * NBIN + NTHR - 1) / NTHR, NTHR, 0, stream>>>(hist1, hist2, cnt);

    dim3 grid(CCLS / CG, H / TY, BATCH);                      // (10, 8, 32)
    k_hist1  <<<grid, NTHR, 0, stream>>>(det, hist1);
    k_select1<<<1, 32, 0, stream>>>(hist1, t1, above1);
    k_hist2  <<<grid, NTHR, 0, stream>>>(det, t1, hist2);
    k_select2<<<1, 32, 0, stream>>>(hist2, t1, above1, thresh);
    k_collect<<<grid, NTHR, 0, stream>>>(det, thresh, cnt, cs, ci);
    k_final  <<<BATCH, NTHR, 0, stream>>>(det, cs, ci, cnt, out);
}